// LocalAttentionBlock_22505628631681
// MI455X (gfx1250) — compile-verified
//
#include <hip/hip_runtime.h>
#include <hip/hip_bf16.h>

// ---------------------------------------------------------------------------
// LocalAttentionBlock for MI455X (gfx1250).
// v4: GEMMs use the Tensor Data Mover (tensor_load_to_lds, 6-arg builtin on
//     this toolchain) to double-buffer A/W tiles into LDS (TENSORcnt-
//     synchronized), feeding V_WMMA_F32_16X16X4_F32 from ds_read_b128
//     fragments. Attention keeps the register-direct WMMA path.
// ---------------------------------------------------------------------------

#define S_LEN 4096
#define DMODEL 512
#define NHEADS 8
#define HDIM 64
#define ROW3 1536   // 3*DMODEL
#define HIDDEN 2048
#define KT_N 17     // key tiles per query block: qb-8 .. qb+8
#define KT_W 272    // 17*16 keys in band window per query block

typedef float v2f __attribute__((ext_vector_type(2)));
typedef float v4f __attribute__((ext_vector_type(4)));
typedef float v8f __attribute__((ext_vector_type(8)));
typedef unsigned int u32x4 __attribute__((ext_vector_type(4)));
typedef int i32x4 __attribute__((ext_vector_type(4)));
typedef int i32x8 __attribute__((ext_vector_type(8)));

// fp32 WMMA: D(16x16) = A(16x4) * B(4x16) + C.
// Per 8-K block loaded as one b128/lane: half-wave 0 owns K-slots {k..k+3},
// half 1 owns {k+4..k+7}; WMMA#1 consumes .xy, WMMA#2 consumes .zw. The K
// permutation is identical for A and B, so the accumulated sum is exact.
__device__ __forceinline__ v8f wmma_f32(v2f a, v2f b, v8f c) {
    return __builtin_amdgcn_wmma_f32_16x16x4_f32(
        false, a, false, b, (short)0, c, false, false);
}
__device__ __forceinline__ v2f lo2(v4f v) { return __builtin_shufflevector(v, v, 0, 1); }
__device__ __forceinline__ v2f hi2(v4f v) { return __builtin_shufflevector(v, v, 2, 3); }

// ---------------------------------------------------------------------------
// TDM: issue one tensor_load_to_lds of a 2-D fp32 tile (32 cols x `rows`)
// from global (row stride `row_stride` elements) into LDS at byte offset
// `lds_off`. D# packing per CDNA5 ISA 8.3/8.4; groups 2/3 and the trailing
// group are zero for a 2-D tile (tile_dim2..4 = 0 => unused).
// ---------------------------------------------------------------------------
__device__ __forceinline__ void tdm_load_tile32(
    unsigned lds_off, const float* gptr, int rows, int row_stride) {
    unsigned long long ga = (unsigned long long)(uintptr_t)gptr;
    unsigned td0 = (unsigned)row_stride;   // tensor_dim0 (elements)
    unsigned td1 = (unsigned)rows;         // tensor_dim1

    u32x4 g0;
    g0[0] = 1u;                                        // count=1, user-mode
    g0[1] = lds_off;                                   // lds_addr (bytes)
    g0[2] = (unsigned)(ga & 0xFFFFFFFFull);            // global_addr[31:0]
    g0[3] = (unsigned)((ga >> 32) & 0x1FFFFFFull)      // global_addr[56:32]
          | (2u << 30);                                // type = 2 ("image")

    i32x8 g1;
    g1[0] = (int)(2u << 16);                           // data_size = 4 bytes
    g1[1] = (int)((td0 & 0xFFFFu) << 16);              // tensor_dim0[15:0]
    g1[2] = (int)(((td0 >> 16) & 0xFFFFu)              // tensor_dim0[31:16]
          | ((td1 & 0xFFFFu) << 16));                  // tensor_dim1[15:0]
    g1[3] = (int)(((td1 >> 16) & 0xFFFFu)              // tensor_dim1[31:16]
          | (32u << 16));                              // tile_dim0 = 32 cols
    g1[4] = (int)(td1 & 0xFFFFu);                      // tile_dim1 = rows; tile_dim2=0
    g1[5] = (int)td0;                                  // tensor_dim0_stride[31:0]
    g1[6] = 0;                                         // stride0[47:32]=0; stride1 lo=0
    g1[7] = 0;

    i32x4 z4 = {0, 0, 0, 0};
    i32x8 z8 = {0, 0, 0, 0, 0, 0, 0, 0};
    __builtin_amdgcn_tensor_load_to_lds(g0, g1, z4, z4, z8, 0);
}

// ---------------------------------------------------------------------------
// RMSNorm: one 256-thread block per row of [S, 512]
// ---------------------------------------------------------------------------
__global__ __launch_bounds__(256) void rmsnorm_kernel(
    const float* __restrict__ x, const float* __restrict__ w,
    float* __restrict__ out) {
    const int row = blockIdx.x;
    const int t = threadIdx.x;
    __shared__ float red[256];
    const float* xr = x + (size_t)row * DMODEL;
    float v0 = xr[t];
    float v1 = xr[t + 256];
    red[t] = v0 * v0 + v1 * v1;
    __syncthreads();
    #pragma unroll
    for (int s2 = 128; s2 > 0; s2 >>= 1) {
        if (t < s2) red[t] += red[t + s2];
        __syncthreads();
    }
    float rs = rsqrtf(red[0] * (1.0f / DMODEL) + 1e-6f);
    out[(size_t)row * DMODEL + t]       = v0 * rs * w[t];
    out[(size_t)row * DMODEL + t + 256] = v1 * rs * w[t + 256];
}

// ---------------------------------------------------------------------------
// One 32-K chunk of WMMA from LDS-staged tiles.
// Ab: 16x32 A tile (row m at m*32); Wb: 64x32 W tile (row n at n*32).
// 20 ds_read_b128 fragments, then 32 WMMAs.
// ---------------------------------------------------------------------------
__device__ __forceinline__ void gemm_chunk(
    const float* Ab, const float* Wb, int lm, int half, v8f* acc) {
    v4f af[4];
    v4f wf[4][4];
    #pragma unroll
    for (int kb = 0; kb < 4; ++kb)
        af[kb] = *(const v4f*)(Ab + lm * 32 + kb * 8 + half * 4);
    #pragma unroll
    for (int kb = 0; kb < 4; ++kb)
        #pragma unroll
        for (int t = 0; t < 4; ++t)
            wf[kb][t] = *(const v4f*)(Wb + (t * 16 + lm) * 32 + kb * 8 + half * 4);
    #pragma unroll
    for (int kb = 0; kb < 4; ++kb)
        #pragma unroll
        for (int t = 0; t < 4; ++t) {
            acc[t] = wmma_f32(lo2(af[kb]), lo2(wf[kb][t]), acc[t]);
            acc[t] = wmma_f32(hi2(af[kb]), hi2(wf[kb][t]), acc[t]);
        }
}

// ---------------------------------------------------------------------------
// WMMA GEMM: C[S,N] = A[S,K] @ W[N,K]^T + bias[N] (+ resid[S,N])
// One wave per 16x64 output strip; grid = (N/64, S/16), block = 32.
// K loop: 32-K chunks staged into double-buffered LDS by the TDM; the wave
// only ever waits for the previous chunk (s_wait_tensorcnt 2) while the next
// chunk's DMA is in flight.
// ---------------------------------------------------------------------------
__global__ __launch_bounds__(32) void gemm16_kernel(
    const float* __restrict__ A, const float* __restrict__ W,
    const float* __restrict__ bias, const float* __restrict__ resid,
    float* __restrict__ C, int K, int N) {
    const int lane = threadIdx.x & 31;
    const int half = lane >> 4;
    const int lm   = lane & 15;
    const int row0 = blockIdx.y * 16;
    const int n0   = blockIdx.x * 64;

    __shared__ __align__(16) float sA[2][16 * 32];   // 2 x 2 KB
    __shared__ __align__(16) float sW[2][64 * 32];   // 2 x 8 KB

    const unsigned oA[2] = { (unsigned)(uintptr_t)&sA[0][0],
                             (unsigned)(uintptr_t)&sA[1][0] };
    const unsigned oW[2] = { (unsigned)(uintptr_t)&sW[0][0],
                             (unsigned)(uintptr_t)&sW[1][0] };

    const float* Arow = A + (size_t)row0 * K;
    const float* Wrow = W + (size_t)n0 * K;
    const int nchunks = K / 32;

    // prologue: DMA chunks 0 and 1
    tdm_load_tile32(oA[0], Arow,      16, K);
    tdm_load_tile32(oW[0], Wrow,      64, K);
    tdm_load_tile32(oA[1], Arow + 32, 16, K);
    tdm_load_tile32(oW[1], Wrow + 32, 64, K);

    v8f acc[4] = {};
    for (int c = 0; c < nchunks - 1; ++c) {
        __builtin_amdgcn_s_wait_tensorcnt(2);   // chunk c resident in LDS
        const int s = c & 1;
        gemm_chunk(sA[s], sW[s], lm, half, acc);
        if (c + 2 < nchunks) {                  // refill just-consumed buffer
            tdm_load_tile32(oA[s], Arow + (c + 2) * 32, 16, K);
            tdm_load_tile32(oW[s], Wrow + (c + 2) * 32, 64, K);
        }
    }
    __builtin_amdgcn_s_wait_tensorcnt(0);
    gemm_chunk(sA[(nchunks - 1) & 1], sW[(nchunks - 1) & 1], lm, half, acc);

    #pragma unroll
    for (int t = 0; t < 4; ++t) {
        const int col = n0 + t * 16 + lm;
        const float bv = bias[col];
        #pragma unroll
        for (int r = 0; r < 8; ++r) {
            const int row = row0 + r + 8 * half;
            float val = acc[t][r] + bv;
            if (resid) val += resid[(size_t)row * N + col];
            C[(size_t)row * N + col] = val;
        }
    }
}

// ---------------------------------------------------------------------------
// Banded attention: one wave per (head, 16-query block).
// Window 128 == 8 tiles of 16 -> key tiles qb-8..qb+8 (17 tiles).
// Q fragments hoisted to registers; scores staged in LDS; in-wave softmax;
// PV via WMMA with probs read from LDS as b128 fragments.
// ---------------------------------------------------------------------------
__global__ __launch_bounds__(32) void attn_kernel(
    const float* __restrict__ qkv, float* __restrict__ attn) {
    const int lane = threadIdx.x & 31;
    const int half = lane >> 4;
    const int lm   = lane & 15;
    const int h  = blockIdx.x & (NHEADS - 1);
    const int qb = blockIdx.x >> 3;

    __shared__ __align__(16) float sc[16 * KT_W];   // 17 KB
    const float scale = 0.125f;                     // 1/sqrt(64)

    const float* qp = qkv + (size_t)(qb * 16 + lm) * ROW3 + h * HDIM + half * 4;
    v4f qf[8];
    #pragma unroll
    for (int i = 0; i < 8; ++i) qf[i] = *(const v4f*)(qp + i * 8);

    // ---- scores: Q(16x64) . K_tile^T(64x16), masked, -> LDS ----
    for (int kt = 0; kt < KT_N; ++kt) {
        const int kr = qb - 8 + kt;
        if (kr < 0 || kr >= S_LEN / 16) {
            #pragma unroll
            for (int r = 0; r < 8; ++r)
                sc[(r + 8 * half) * KT_W + kt * 16 + lm] = -1e30f;
            continue;
        }
        const float* kp = qkv + (size_t)(kr * 16 + lm) * ROW3 + DMODEL + h * HDIM + half * 4;
        v8f c = {};
        #pragma unroll
        for (int i = 0; i < 8; ++i) {
            v4f kf = *(const v4f*)(kp + i * 8);
            c = wmma_f32(lo2(qf[i]), lo2(kf), c);
            c = wmma_f32(hi2(qf[i]), hi2(kf), c);
        }
        #pragma unroll
        for (int r = 0; r < 8; ++r) {
            const int i = qb * 16 + r + 8 * half;   // query index
            const int j = kr * 16 + lm;             // key index
            int d = i - j; d = d < 0 ? -d : d;
            float val = c[r] * scale;
            sc[(r + 8 * half) * KT_W + kt * 16 + lm] = (d > 128) ? -1e30f : val;
        }
    }
    __syncthreads();

    // ---- softmax: lane handles row lm, half-wave splits the 272 keys ----
    {
        float* rowp = sc + lm * KT_W + half * (KT_W / 2);
        float mx = -1e30f;
        for (int k = 0; k < KT_W / 2; ++k) mx = fmaxf(mx, rowp[k]);
        mx = fmaxf(mx, __shfl_xor(mx, 16, 32));
        float sum = 0.0f;
        for (int k = 0; k < KT_W / 2; ++k) {
            float e = __expf(rowp[k] - mx);
            rowp[k] = e;
            sum += e;
        }
        sum += __shfl_xor(sum, 16, 32);
        const float inv = 1.0f / sum;
        for (int k = 0; k < KT_W / 2; ++k) rowp[k] *= inv;
    }
    __syncthreads();

    // ---- PV: attn(16x64) = probs(16x272) . V_window(272x64) ----
    const int kstart = (qb - 8) * 16;
    const float* vbase = qkv + 2 * DMODEL + h * HDIM + lm;
    v8f accs[4] = {};
    for (int k0 = 0; k0 < KT_W; k0 += 8) {
        v4f a = *(const v4f*)(sc + lm * KT_W + k0 + half * 4);
        const int kr0 = kstart + k0 + half * 4;
        int r0 = kr0;     r0 = r0 < 0 ? 0 : (r0 >= S_LEN ? S_LEN - 1 : r0);
        int r1 = kr0 + 1; r1 = r1 < 0 ? 0 : (r1 >= S_LEN ? S_LEN - 1 : r1);
        int r2 = kr0 + 2; r2 = r2 < 0 ? 0 : (r2 >= S_LEN ? S_LEN - 1 : r2);
        int r3 = kr0 + 3; r3 = r3 < 0 ? 0 : (r3 >= S_LEN ? S_LEN - 1 : r3);
        const float* p0 = vbase + (size_t)r0 * ROW3;
        const float* p1 = vbase + (size_t)r1 * ROW3;
        const float* p2 = vbase + (size_t)r2 * ROW3;
        const float* p3 = vbase + (size_t)r3 * ROW3;
        #pragma unroll
        for (int t = 0; t < 4; ++t) {
            v2f bl = { p0[t * 16], p1[t * 16] };   // K-slots {kr0, kr0+1}
            v2f bh = { p2[t * 16], p3[t * 16] };   // K-slots {kr0+2, kr0+3}
            accs[t] = wmma_f32(lo2(a), bl, accs[t]);
            accs[t] = wmma_f32(hi2(a), bh, accs[t]);
        }
    }
    #pragma unroll
    for (int t = 0; t < 4; ++t)
        #pragma unroll
        for (int r = 0; r < 8; ++r)
            attn[(size_t)(qb * 16 + r + 8 * half) * DMODEL + h * HDIM + t * 16 + lm] = accs[t][r];
}

// ---------------------------------------------------------------------------
// SwiGLU combine: a = silu(a) * b
// ---------------------------------------------------------------------------
__global__ __launch_bounds__(256) void silu_mul_kernel(
    float* __restrict__ a, const float* __restrict__ b, int n) {
    int i = blockIdx.x * blockDim.x + threadIdx.x;
    if (i < n) {
        float x = a[i];
        float s = x / (1.0f + __expf(-x));
        a[i] = s * b[i];
    }
}

// ---------------------------------------------------------------------------
// Workspace layout (floats); ~80 MB total. h1 overlays dead qkv+attn regions.
// ---------------------------------------------------------------------------
#define OFF_XN   ((size_t)0)                        // 4096*512
#define OFF_QKV  ((size_t)2097152)                  // 4096*1536
#define OFF_ATTN ((size_t)8388608)                  // 4096*512
#define OFF_X1   ((size_t)10485760)                 // 4096*512
#define OFF_H1   OFF_QKV                            // 4096*2048 (reuse)
#define OFF_H2   ((size_t)12582912)                 // 4096*2048

extern "C" void kernel_launch(void* const* d_in, const int* in_sizes, int n_in,
                              void* d_out, int out_size, void* d_ws, size_t ws_size,
                              hipStream_t stream) {
    const float* x    = (const float*)d_in[0];
    const float* inw  = (const float*)d_in[1];
    const float* inb  = (const float*)d_in[2];
    const float* outw = (const float*)d_in[3];
    const float* outb = (const float*)d_in[4];
    const float* n1w  = (const float*)d_in[5];
    const float* n2w  = (const float*)d_in[6];
    const float* w1w  = (const float*)d_in[7];
    const float* w1b  = (const float*)d_in[8];
    const float* w3w  = (const float*)d_in[9];
    const float* w3b  = (const float*)d_in[10];
    const float* w2w  = (const float*)d_in[11];
    const float* w2b  = (const float*)d_in[12];
    float* out = (float*)d_out;

    float* ws   = (float*)d_ws;
    float* xn   = ws + OFF_XN;
    float* qkv  = ws + OFF_QKV;
    float* attn = ws + OFF_ATTN;
    float* x1   = ws + OFF_X1;
    float* h1   = ws + OFF_H1;
    float* h2   = ws + OFF_H2;

    // 1) xn = rmsnorm(x, norm1_w)
    rmsnorm_kernel<<<S_LEN, 256, 0, stream>>>(x, n1w, xn);

    // 2) qkv = xn @ in_proj_w^T + in_proj_b        [S, 1536]
    gemm16_kernel<<<dim3(ROW3 / 64, S_LEN / 16), 32, 0, stream>>>(
        xn, inw, inb, nullptr, qkv, DMODEL, ROW3);

    // 3) banded multi-head attention -> attn [S, 512]
    attn_kernel<<<dim3((S_LEN / 16) * NHEADS), 32, 0, stream>>>(qkv, attn);

    // 4) x1 = x + attn @ out_proj_w^T + out_proj_b
    gemm16_kernel<<<dim3(DMODEL / 64, S_LEN / 16), 32, 0, stream>>>(
        attn, outw, outb, x, x1, DMODEL, DMODEL);

    // 5) xn = rmsnorm(x1, norm2_w)
    rmsnorm_kernel<<<S_LEN, 256, 0, stream>>>(x1, n2w, xn);

    // 6) h1 = xn @ w1^T + b1 ; h2 = xn @ w3^T + b3     [S, 2048]
    gemm16_kernel<<<dim3(HIDDEN / 64, S_LEN / 16), 32, 0, stream>>>(
        xn, w1w, w1b, nullptr, h1, DMODEL, HIDDEN);
    gemm16_kernel<<<dim3(HIDDEN / 64, S_LEN / 16), 32, 0, stream>>>(
        xn, w3w, w3b, nullptr, h2, DMODEL, HIDDEN);

    // 7) h1 = silu(h1) * h2
    int n = S_LEN * HIDDEN;
    silu_mul_kernel<<<(n + 255) / 256, 256, 0, stream>>>(h1, h2, n);

    // 8) out = x1 + h1 @ w2^T + b2
    gemm16_kernel<<<dim3(DMODEL / 64, S_LEN / 16), 32, 0, stream>>>(
        h1, w2w, w2b, x1, out, HIDDEN, DMODEL);
}